// GNN_Infograph_75728863363725
// MI455X (gfx1250) — compile-verified
//
#include <hip/hip_runtime.h>
#include <hip/hip_bf16.h>

typedef __attribute__((ext_vector_type(2))) float v2f;
typedef __attribute__((ext_vector_type(8))) float v8f;

#define DFEAT 128
#define WSTR  136   // padded LDS row stride for weight matrices (bank-conflict free B-frag reads)
#define HSTR  132   // padded LDS row stride for per-wave H tile
#define WAVES_PER_BLOCK 8

// ---------------- utility kernels ----------------

__global__ void zero_f32(float* __restrict__ p, long n) {
    long i = (long)blockIdx.x * blockDim.x + threadIdx.x;
    if (i < n) p[i] = 0.0f;
}

__global__ void count_nodes(const int* __restrict__ batch, float* __restrict__ counts, int n) {
    int i = blockIdx.x * blockDim.x + threadIdx.x;
    if (i < n) atomicAdd(&counts[batch[i]], 1.0f);
}

__global__ void scale_pool(float* __restrict__ pool, const float* __restrict__ counts,
                           int G, int LD) {
    long i = (long)blockIdx.x * blockDim.x + threadIdx.x;
    if (i < (long)G * LD) {
        int g = (int)(i / LD);
        pool[i] /= fmaxf(counts[g], 1.0f);
    }
}

// One wave (32 lanes) per edge: lane handles 4 consecutive features (32*4 = 128).
__global__ void scatter_add(const float* __restrict__ xprev, long xstride,
                            const int* __restrict__ src, const int* __restrict__ dst,
                            float* __restrict__ agg, int E) {
    long t = (long)blockIdx.x * blockDim.x + threadIdx.x;
    int e = (int)(t >> 5);
    if (e >= E) return;
    int c = ((int)t & 31) << 2;
    int s = src[e], d = dst[e];
    const float4 v = *(const float4*)(xprev + (long)s * xstride + c);
    float* out = agg + (long)d * DFEAT + c;
    atomicAdd(out + 0, v.x);
    atomicAdd(out + 1, v.y);
    atomicAdd(out + 2, v.z);
    atomicAdd(out + 3, v.w);
}

// ---------------- fused GIN MLP (WMMA f32 16x16x4) ----------------
// Per block: stage W1, W2, b1, b2 in LDS. Each of 8 waves owns a 16-row node
// tile: H = (1+eps)*x + agg -> T = relu(H@W1+b1) (written over H tile in LDS)
// -> X = relu(T@W2+b2) stored to xs output + pooled via atomics.

__global__ __launch_bounds__(256)
void gin_mlp(const float* __restrict__ xprev, long xstride,
             const float* __restrict__ agg,
             const float* __restrict__ W1g, const float* __restrict__ b1g,
             const float* __restrict__ W2g, const float* __restrict__ b2g,
             const float* __restrict__ epsv, int layer,
             const int* __restrict__ batch,
             float* __restrict__ xs_out,    // pre-offset by layer*DFEAT, row stride LD
             float* __restrict__ pool_out,  // pre-offset by layer*DFEAT, row stride LD
             int N, int LD) {
    extern __shared__ float smem[];
    float* W1l = smem;                   // [128 * WSTR]
    float* W2l = W1l + DFEAT * WSTR;     // [128 * WSTR]
    float* b1l = W2l + DFEAT * WSTR;     // [128]
    float* b2l = b1l + DFEAT;            // [128]
    float* Hall = b2l + DFEAT;           // [WAVES_PER_BLOCK * 16 * HSTR]

    const int tid = threadIdx.x;

    // cooperative weight staging (row-major [k][n], padded stride)
    for (int idx = tid; idx < DFEAT * DFEAT; idx += 256) {
        int r = idx >> 7, c = idx & 127;
        W1l[r * WSTR + c] = W1g[idx];
        W2l[r * WSTR + c] = W2g[idx];
    }
    if (tid < DFEAT) { b1l[tid] = b1g[tid]; b2l[tid] = b2g[tid]; }
    __syncthreads();

    const int wave = tid >> 5;
    const int lane = tid & 31;
    float* H = Hall + wave * (16 * HSTR);
    const long tile_base = ((long)blockIdx.x * WAVES_PER_BLOCK + wave) * 16;

    const float onepe = 1.0f + epsv[layer];

    // build H tile = (1+eps)*x + agg  (rows past N treated as zero)
    for (int idx = lane; idx < 16 * DFEAT; idx += 32) {
        int r = idx >> 7, c = idx & 127;
        long node = tile_base + r;
        float v = 0.0f;
        if (node < N) v = onepe * xprev[node * xstride + c] + agg[node * (long)DFEAT + c];
        H[r * HSTR + c] = v;
    }

    // WMMA f32 16x16x4 per-lane fragment geometry
    const int rowa    = lane & 15;           // A: matrix row
    const int khalf   = (lane >> 4) << 1;    // A/B: K sub-offset 0 or 2
    const int coln    = lane & 15;           // B/D: column within 16-tile
    const int rowbase = (lane >> 4) << 3;    // D: rows 0..7 or 8..15

    // ---------------- GEMM1: T = relu(H @ W1 + b1) ----------------
    v2f afrag[32];
#pragma unroll
    for (int k = 0; k < 32; ++k)
        afrag[k] = *(const v2f*)(H + rowa * HSTR + 4 * k + khalf);

#pragma unroll 1
    for (int n = 0; n < 8; ++n) {
        v8f acc = {};
#pragma unroll
        for (int k = 0; k < 32; ++k) {
            const float* wp = W1l + (4 * k + khalf) * WSTR + n * 16 + coln;
            v2f b;
            b.x = wp[0];
            b.y = wp[WSTR];
            acc = __builtin_amdgcn_wmma_f32_16x16x4_f32(
                false, afrag[k], false, b, (short)0, acc, false, false);
        }
        float bias = b1l[n * 16 + coln];
#pragma unroll
        for (int j = 0; j < 8; ++j) {
            float v = acc[j] + bias;
            v = fmaxf(v, 0.0f);
            H[(rowbase + j) * HSTR + n * 16 + coln] = v;  // overwrite H with T
        }
    }

    // ---------------- GEMM2: X = relu(T @ W2 + b2) ----------------
#pragma unroll
    for (int k = 0; k < 32; ++k)
        afrag[k] = *(const v2f*)(H + rowa * HSTR + 4 * k + khalf);

#pragma unroll 1
    for (int n = 0; n < 8; ++n) {
        v8f acc = {};
#pragma unroll
        for (int k = 0; k < 32; ++k) {
            const float* wp = W2l + (4 * k + khalf) * WSTR + n * 16 + coln;
            v2f b;
            b.x = wp[0];
            b.y = wp[WSTR];
            acc = __builtin_amdgcn_wmma_f32_16x16x4_f32(
                false, afrag[k], false, b, (short)0, acc, false, false);
        }
        float bias = b2l[n * 16 + coln];
#pragma unroll
        for (int j = 0; j < 8; ++j) {
            long node = tile_base + rowbase + j;
            if (node < N) {
                float v = fmaxf(acc[j] + bias, 0.0f);
                int col = n * 16 + coln;
                xs_out[node * LD + col] = v;
                atomicAdd(&pool_out[(long)batch[node] * LD + col], v);
            }
        }
    }
}

// ---------------- host-side orchestration ----------------

extern "C" void kernel_launch(void* const* d_in, const int* in_sizes, int n_in,
                              void* d_out, int out_size, void* d_ws, size_t ws_size,
                              hipStream_t stream) {
    const float* x     = (const float*)d_in[0];
    const int*   edge  = (const int*)d_in[1];
    const int*   batch = (const int*)d_in[2];
    const float* W1    = (const float*)d_in[3];
    const float* b1    = (const float*)d_in[4];
    const float* W2    = (const float*)d_in[5];
    const float* b2    = (const float*)d_in[6];
    const float* eps   = (const float*)d_in[7];

    const int N  = in_sizes[0] / DFEAT;
    const int E  = in_sizes[1] / 2;
    const int L  = in_sizes[3] / (DFEAT * DFEAT);
    const int LD = L * DFEAT;
    const int G  = (int)(((long)out_size - (long)N * LD) / LD);

    float* pool = (float*)d_out;              // [G, LD]
    float* xs   = pool + (size_t)G * LD;      // [N, LD]

    float* counts = (float*)d_ws;                          // [G]
    float* agg    = (float*)((char*)d_ws + 8192);          // [N, DFEAT]

    const int* srcIdx = edge;
    const int* dstIdx = edge + E;

    const int TB = 256;
    auto blocks = [&](long n) { return (unsigned)((n + TB - 1) / TB); };

    // zero counts + pool accumulators
    zero_f32<<<blocks(G), TB, 0, stream>>>(counts, G);
    zero_f32<<<blocks((long)G * LD), TB, 0, stream>>>(pool, (long)G * LD);
    count_nodes<<<blocks(N), TB, 0, stream>>>(batch, counts, N);

    const size_t smem_bytes =
        (size_t)(2 * DFEAT * WSTR + 2 * DFEAT + WAVES_PER_BLOCK * 16 * HSTR) * sizeof(float);
    const unsigned mlp_blocks =
        (unsigned)((N + 16 * WAVES_PER_BLOCK - 1) / (16 * WAVES_PER_BLOCK));

    for (int i = 0; i < L; ++i) {
        const float* xprev  = (i == 0) ? x : (xs + (size_t)(i - 1) * DFEAT);
        long         stride = (i == 0) ? DFEAT : LD;

        zero_f32<<<blocks((long)N * DFEAT), TB, 0, stream>>>(agg, (long)N * DFEAT);
        scatter_add<<<blocks((long)E * 32), TB, 0, stream>>>(xprev, stride, srcIdx, dstIdx,
                                                             agg, E);
        gin_mlp<<<mlp_blocks, TB, smem_bytes, stream>>>(
            xprev, stride, agg,
            W1 + (size_t)i * DFEAT * DFEAT, b1 + (size_t)i * DFEAT,
            W2 + (size_t)i * DFEAT * DFEAT, b2 + (size_t)i * DFEAT,
            eps, i, batch,
            xs + (size_t)i * DFEAT, pool + (size_t)i * DFEAT,
            N, LD);
    }

    scale_pool<<<blocks((long)G * LD), TB, 0, stream>>>(pool, counts, G, LD);
}